// RocketKernel_8598524526908
// MI455X (gfx1250) — compile-verified
//
#include <hip/hip_runtime.h>
#include <hip/hip_bf16.h>

#define B_N   32
#define T_N   512
#define K_N   1000
#define LMAXN 11

typedef __attribute__((ext_vector_type(2))) float v2f;
typedef __attribute__((ext_vector_type(8))) float v8f;
typedef __attribute__((ext_vector_type(4))) int   v4i;

typedef __attribute__((address_space(1))) v4i* gv4i_p;  // global (AS1)
typedef __attribute__((address_space(3))) v4i* lv4i_p;  // LDS (AS3)

#if defined(__has_builtin)
#if __has_builtin(__builtin_amdgcn_global_load_async_to_lds_b128)
#define USE_ASYNC_LDS 1
#endif
#endif

static __device__ __forceinline__ int imin(int a, int b) { return a < b ? a : b; }

static __device__ __forceinline__ void rk_wait_async() {
#if defined(USE_ASYNC_LDS)
#if __has_builtin(__builtin_amdgcn_s_wait_asynccnt)
  __builtin_amdgcn_s_wait_asynccnt(0);
#else
  asm volatile("s_wait_asynccnt 0" ::: "memory");
#endif
#endif
}

// ---------------------------------------------------------------------------
// Pass 1: stable rank-sort of kernels by (dilation, padding) key.
// perm[rank] = kernel index; kernels with equal key become contiguous.
// O(K^2) = 1M int compares total -- negligible, fully deterministic.
// ---------------------------------------------------------------------------
__global__ __launch_bounds__(256) void rk_rank_sort(const int* __restrict__ dil,
                                                    const int* __restrict__ pad,
                                                    int* __restrict__ perm) {
  int k = blockIdx.x * blockDim.x + threadIdx.x;
  if (k >= K_N) return;
  int key = (dil[k] << 16) | pad[k];
  int rank = 0;
  for (int j = 0; j < K_N; ++j) {
    int kj = (dil[j] << 16) | pad[j];
    rank += (kj < key) || (kj == key && j < k);
  }
  perm[rank] = k;
}

// ---------------------------------------------------------------------------
// Pass 2: scan the sorted order, emit chunks of <=16 kernels that share one
// (dilation, padding) key. Single thread, K iterations.
// ---------------------------------------------------------------------------
__global__ void rk_build_chunks(const int* __restrict__ dil,
                                const int* __restrict__ pad,
                                const int* __restrict__ perm,
                                int* __restrict__ cstart,
                                int* __restrict__ csize,
                                int* __restrict__ ncp) {
  if (threadIdx.x != 0 || blockIdx.x != 0) return;
  int nc = 0, i = 0;
  while (i < K_N) {
    int k0 = perm[i];
    int key = (dil[k0] << 16) | pad[k0];
    int e = i + 1;
    while (e < K_N) {
      int ke = perm[e];
      if ((((dil[ke]) << 16) | pad[ke]) != key) break;
      ++e;
    }
    for (int s = i; s < e; s += 16) {
      cstart[nc] = s;
      csize[nc] = imin(16, e - s);
      ++nc;
    }
    i = e;
  }
  *ncp = nc;
}

// ---------------------------------------------------------------------------
// Pass 3: one wave32 per (chunk, batch). Stage X row in LDS (async-to-LDS on
// gfx1250 when available), then for each pair of t-tiles run two independent
// chains of 3x v_wmma_f32_16x16x4_f32 against the chunk's weight fragment B
// (bias preloaded in C). PPV count and max accumulate in registers.
// ---------------------------------------------------------------------------
__global__ __launch_bounds__(32) void rk_conv_wmma(
    const float* __restrict__ X, const float* __restrict__ W,
    const float* __restrict__ biases, const int* __restrict__ lens,
    const int* __restrict__ dils, const int* __restrict__ pads,
    const int* __restrict__ perm, const int* __restrict__ cstart,
    const int* __restrict__ csize, const int* __restrict__ ncp,
    float* __restrict__ out) {
  int c = blockIdx.x;
  if (c >= *ncp) return;          // uniform across the wave
  int b = blockIdx.y;
  int start = cstart[c];
  int size  = csize[c];
  int lane  = threadIdx.x;
  int n     = lane & 15;          // D/B column = kernel slot within chunk
  int half  = lane >> 4;

  // ---- stage X[b, :] (2KB) into LDS ----
  __shared__ float sX[T_N];
  const float* xrow = X + b * T_N;
#if defined(USE_ASYNC_LDS)
#pragma unroll
  for (int i = 0; i < 4; ++i) {
    int off = (lane + i * 32) * 4;
    __builtin_amdgcn_global_load_async_to_lds_b128(
        (gv4i_p)(xrow + off), (lv4i_p)&sX[off], 0, 0);
  }
  rk_wait_async();
#else
#pragma unroll
  for (int i = 0; i < 4; ++i) {
    int off = (lane + i * 32) * 4;
    float4 v = *reinterpret_cast<const float4*>(xrow + off);
    *reinterpret_cast<float4*>(&sX[off]) = v;
  }
#endif
  __syncthreads();

  bool active = n < size;
  int k = perm[start + (active ? n : 0)];
  int d = dils[k];                // uniform within chunk by construction
  int p = pads[k];                // uniform within chunk by construction
  int len = lens[k];              // may differ per column (ragged lengths)
  int out_len = T_N + 2 * p - d * (len - 1);   // >= 1 by construction
  float bias = active ? biases[k] : 0.f;

  // ---- B fragments (4x16 taps-by-kernels), per ISA lane layout:
  //      lanes 0-15 hold tap rows {0,1}, lanes 16-31 hold {2,3} ----
  v2f bw[3];
#pragma unroll
  for (int jb = 0; jb < 3; ++jb) {
#pragma unroll
    for (int r = 0; r < 2; ++r) {
      int tap = jb * 4 + 2 * half + r;
      bw[jb][r] = (active && tap < LMAXN) ? W[k * LMAXN + tap] : 0.f;
    }
  }

  // ---- hoist per-tap gather bases: idx = base[jb][r] + t0 ----
  int  tA  = n;                   // A-matrix row M = lane & 15
  int  jA0 = 2 * half;            // tap pair selected by lane half
  int  base[3][2];
  bool tapok[3][2];
#pragma unroll
  for (int jb = 0; jb < 3; ++jb) {
#pragma unroll
    for (int r = 0; r < 2; ++r) {
      int tap = jb * 4 + jA0 + r;
      base[jb][r]  = tA - p + tap * d;
      tapok[jb][r] = tap < LMAXN;
    }
  }

  const float NEGINF = -__builtin_inff();
  float cnt = 0.f;
  float mx  = NEGINF;

  for (int tile = 0; tile < T_N / 16; tile += 2) {
    int t0 = tile * 16;
    int t1 = t0 + 16;
    v8f acc0, acc1;
#pragma unroll
    for (int r = 0; r < 8; ++r) { acc0[r] = bias; acc1[r] = bias; }

#pragma unroll
    for (int jb = 0; jb < 3; ++jb) {
      v2f a0, a1;
#pragma unroll
      for (int r = 0; r < 2; ++r) {
        int i0 = base[jb][r] + t0;
        int i1 = base[jb][r] + t1;
        bool ok0 = tapok[jb][r] & (i0 >= 0) & (i0 < T_N);
        bool ok1 = tapok[jb][r] & (i1 >= 0) & (i1 < T_N);
        int c0 = i0 < 0 ? 0 : (i0 > T_N - 1 ? T_N - 1 : i0);
        int c1 = i1 < 0 ? 0 : (i1 > T_N - 1 ? T_N - 1 : i1);
        float v0 = sX[c0];                       // ds_load_b32 gathers
        float v1 = sX[c1];
        a0[r] = ok0 ? v0 : 0.f;
        a1[r] = ok1 ? v1 : 0.f;
      }
      // two independent accumulation chains keep the matrix pipe busy
      acc0 = __builtin_amdgcn_wmma_f32_16x16x4_f32(false, a0, false, bw[jb],
                                                   (short)0, acc0, false, false);
      acc1 = __builtin_amdgcn_wmma_f32_16x16x4_f32(false, a1, false, bw[jb],
                                                   (short)0, acc1, false, false);
    }

    // D rows this lane holds: M = r + 8*half, column n
#pragma unroll
    for (int r = 0; r < 8; ++r) {
      int t = t0 + r + 8 * half;
      bool in0 = t < out_len;
      bool in1 = (t + 16) < out_len;
      float v0 = acc0[r];
      float v1 = acc1[r];
      cnt += (in0 && v0 > 0.f) ? 1.f : 0.f;
      cnt += (in1 && v1 > 0.f) ? 1.f : 0.f;
      mx = fmaxf(mx, in0 ? v0 : NEGINF);
      mx = fmaxf(mx, in1 ? v1 : NEGINF);
    }
  }

  // lanes n and n+16 hold complementary row halves of the same column
  cnt += __shfl_xor(cnt, 16, 32);
  mx = fmaxf(mx, __shfl_xor(mx, 16, 32));

  if (half == 0 && active) {
    out[b * (2 * K_N) + 2 * k + 0] = cnt / (float)out_len;  // ppv
    out[b * (2 * K_N) + 2 * k + 1] = mx;                    // max
  }
}

// ---------------------------------------------------------------------------
extern "C" void kernel_launch(void* const* d_in, const int* in_sizes, int n_in,
                              void* d_out, int out_size, void* d_ws, size_t ws_size,
                              hipStream_t stream) {
  const float* X      = (const float*)d_in[0];
  const float* Wt     = (const float*)d_in[1];
  const float* biases = (const float*)d_in[2];
  const int* lens     = (const int*)d_in[3];
  const int* dils     = (const int*)d_in[4];
  const int* pads     = (const int*)d_in[5];
  float* out = (float*)d_out;

  // ws layout: perm[K] | cstart[K] | csize[K] | n_chunks  (~12 KB)
  int* perm   = (int*)d_ws;
  int* cstart = perm + K_N;
  int* csize  = cstart + K_N;
  int* ncp    = csize + K_N;

  rk_rank_sort<<<(K_N + 255) / 256, 256, 0, stream>>>(dils, pads, perm);
  rk_build_chunks<<<1, 32, 0, stream>>>(dils, pads, perm, cstart, csize, ncp);
  rk_conv_wmma<<<dim3(K_N, B_N), 32, 0, stream>>>(X, Wt, biases, lens, dils,
                                                  pads, perm, cstart, csize,
                                                  ncp, out);
}